// CRLoss_56487409877004
// MI455X (gfx1250) — compile-verified
//
#include <hip/hip_runtime.h>
#include <hip/hip_bf16.h>
#include <stdint.h>

#define BSZ 16
#define CH  3
#define H   512
#define W   512
#define HW  (H*W)        // 262144 = 2^18
#define CHW (CH*HW)      // 786432
#define KSZ 15
#define PAD 7
#define NUMPX 262        // 262144/1000

typedef __attribute__((address_space(3))) float lds_f32;

__device__ __forceinline__ float fmin3(float a, float b, float c) {
    return fminf(a, fminf(b, c));
}

// ---------------- Kernel 1: dark channel (min over C) + horizontal 15-min ----
// The three raw channel rows are staged into LDS with gfx1250 async
// global->LDS copies (GLOBAL_LOAD_ASYNC_TO_LDS_B32, tracked by ASYNCcnt),
// then the dark-channel row and the horizontal erosion are computed from LDS.
__global__ void __launch_bounds__(256) k_hmin(const float* __restrict__ foggy,
                                              float* __restrict__ hmin) {
    const int y = blockIdx.x;      // 0..511
    const int b = blockIdx.y;      // 0..15
    const int t = threadIdx.x;     // 0..255
    __shared__ float chbuf[3][W];          // raw channel rows
    __shared__ float row[W + 2*PAD];       // dark-channel row with halo
    const float* fb = foggy + (size_t)b * CHW + (size_t)y * W;

    // Per-lane async copies: each lane copies foggy pixels of row y from global memory into chbuf in LDS
    #pragma unroll
    for (int c = 0; c < 3; ++c) {
        #pragma unroll
        for (int k = 0; k < 2; ++k) {
            const int i = t + k * 256;
            const float* gp = fb + c * HW + i;
            unsigned int ldsoff =
                (unsigned int)(unsigned long long)(lds_f32*)&chbuf[c][i];
            asm volatile("global_load_async_to_lds_b32 %0, %1, off"
                         :: "v"(ldsoff), "v"(gp) : "memory");
        }
    }
    asm volatile("s_wait_asynccnt 0x0" ::: "memory");
    __syncthreads();

    for (int i = t; i < W + 2*PAD; i += 256) {
        int x = i - PAD;
        float v = 1.0f;  // pad value per reference
        if (x >= 0 && x < W) v = fmin3(chbuf[0][x], chbuf[1][x], chbuf[2][x]);
        row[i] = v;
    }
    __syncthreads();

    float* out = hmin + (size_t)b * HW + (size_t)y * W;
    for (int x = t; x < W; x += 256) {
        float m = row[x];
        #pragma unroll
        for (int j = 1; j < KSZ; ++j) m = fminf(m, row[x + j]);
        out[x] = m;
    }
}

// ---------------- Kernel 2: vertical 15-min -> dark ------------------------
__global__ void k_vmin(const float* __restrict__ hmin, float* __restrict__ dark) {
    int idx = blockIdx.x * blockDim.x + threadIdx.x;
    if (idx >= BSZ * HW) return;
    int b = idx >> 18;
    int r = idx & (HW - 1);
    int y = r >> 9;
    int x = r & (W - 1);
    const float* hb = hmin + (size_t)b * HW;
    float m = 1.0f;
    #pragma unroll
    for (int j = -PAD; j <= PAD; ++j) {
        int yy = y + j;
        if (yy >= 0 && yy < H) m = fminf(m, hb[yy * W + x]);
    }
    dark[idx] = m;
}

// ---------------- Kernel 3: exact top-262 by (value,index) key, build A ----
// One 1024-thread block per batch image. 8-pass radix select on 64-bit key
// (float bits << 32) | index; keys unique -> exactly 262 candidates. Bitonic
// sort in LDS gives canonical order => deterministic summation. Reference
// drops the min-key element of the top-262 and divides by 262.
__global__ void __launch_bounds__(1024) k_select(const float* __restrict__ dark,
                                                 const float* __restrict__ foggy,
                                                 float* __restrict__ An) {
    const int b   = blockIdx.x;
    const int tid = threadIdx.x;
    const float* db = dark + (size_t)b * HW;

    __shared__ unsigned int hist[256];
    __shared__ int bstar_s;
    __shared__ unsigned int Ks;
    __shared__ unsigned long long cand[512];
    __shared__ int cnt;

    unsigned long long prefix = 0ull, mask = 0ull;
    unsigned int K = NUMPX;

    for (int pass = 7; pass >= 0; --pass) {
        for (int i = tid; i < 256; i += 1024) hist[i] = 0u;
        __syncthreads();
        const int shift = pass * 8;
        for (int i = tid; i < HW; i += 1024) {
            unsigned long long key =
                ((unsigned long long)__float_as_uint(db[i]) << 32) | (unsigned int)i;
            if ((key & mask) == prefix)
                atomicAdd(&hist[(unsigned int)((key >> shift) & 255ull)], 1u);
        }
        __syncthreads();
        if (tid == 0) {
            unsigned int cum = 0; int bstar = 0; unsigned int Krem = K;
            for (int bb = 255; bb >= 0; --bb) {
                unsigned int c = hist[bb];
                if (cum + c >= K) { bstar = bb; Krem = K - cum; break; }
                cum += c;
            }
            bstar_s = bstar; Ks = Krem;
        }
        __syncthreads();
        prefix |= ((unsigned long long)bstar_s) << shift;
        mask   |= (0xFFull << shift);
        K = Ks;
        __syncthreads();
    }

    // Collect all keys >= threshold key (exactly NUMPX of them).
    if (tid == 0) cnt = 0;
    __syncthreads();
    for (int i = tid; i < HW; i += 1024) {
        unsigned long long key =
            ((unsigned long long)__float_as_uint(db[i]) << 32) | (unsigned int)i;
        if (key >= prefix) {
            int p = atomicAdd(&cnt, 1);
            if (p < 512) cand[p] = key;
        }
    }
    __syncthreads();
    int m = cnt < 512 ? cnt : 512;
    for (int i = tid; i < 512; i += 1024) if (i >= m) cand[i] = 0ull;
    __syncthreads();

    // Bitonic ascending sort of 512 keys (threads 0..511 own one slot each).
    for (int k = 2; k <= 512; k <<= 1) {
        for (int j = k >> 1; j > 0; j >>= 1) {
            if (tid < 512) {
                int i = tid, ixj = i ^ j;
                if (ixj > i) {
                    unsigned long long a = cand[i], bb = cand[ixj];
                    bool up = ((i & k) == 0);
                    if ((a > bb) == up) { cand[i] = bb; cand[ixj] = a; }
                }
            }
            __syncthreads();
        }
    }

    // Top-262 at positions 250..511; drop position 250; sum 251..511; /262.
    if (tid == 0) {
        const float* fb = foggy + (size_t)b * CHW;
        float s0 = 0.f, s1 = 0.f, s2 = 0.f;
        for (int p = 512 - NUMPX + 1; p < 512; ++p) {   // 251..511 (261 elems)
            unsigned int ii = (unsigned int)(cand[p] & 0xFFFFFFFFull);
            s0 += fb[ii];
            s1 += fb[ii + HW];
            s2 += fb[ii + 2*HW];
        }
        s0 /= (float)NUMPX; s1 /= (float)NUMPX; s2 /= (float)NUMPX;
        float t = fmaxf(s0 + s1 + s2, 1e-6f);
        An[b*3+0] = s0 / t;
        An[b*3+1] = s1 / t;
        An[b*3+2] = s2 / t;
    }
}

// ---------------- Kernel 4: elementwise loss -> per-block partial sums ------
__global__ void __launch_bounds__(256) k_loss(const float* __restrict__ clear_,
                                              const float* __restrict__ foggy,
                                              const float* __restrict__ An,
                                              float* __restrict__ partials) {
    const int t = threadIdx.x;
    const int base = blockIdx.x * (256 * 8);   // 2048 pixels/block, same batch image
    const int b = base >> 18;
    const float A0 = An[b*3+0], A1 = An[b*3+1], A2 = An[b*3+2];
    const float* cb = clear_ + (size_t)b * CHW;
    const float* fb = foggy  + (size_t)b * CHW;
    const int rbase = base & (HW - 1);
    float acc = 0.0f;
    #pragma unroll
    for (int kk = 0; kk < 8; ++kk) {
        int r = rbase + kk * 256 + t;
        // gfx1250 prefetch of the next tile (emits global_prefetch_b8)
        __builtin_prefetch(&fb[r + 2048], 0, 1);
        __builtin_prefetch(&cb[r + 2048], 0, 1);
        float c0 = cb[r], c1 = cb[r + HW], c2 = cb[r + 2*HW];
        float f0 = fb[r], f1 = fb[r + HW], f2 = fb[r + 2*HW];
        float sc = fmaxf(c0 + c1 + c2, 1e-6f);
        float sf = fmaxf(f0 + f1 + f2, 1e-6f);
        float g0 = c0/sc, g1 = c1/sc, g2 = c2/sc;
        float s0 = f0/sf, s1 = f1/sf, s2 = f2/sf;
        float fd0 = s0-A0, fd1 = s1-A1, fd2 = s2-A2;
        float cd0 = g0-A0, cd1 = g1-A1, cd2 = g2-A2;
        float nf = fmaxf(sqrtf(fd0*fd0 + fd1*fd1 + fd2*fd2), 1e-12f);
        float nc = fmaxf(sqrtf(cd0*cd0 + cd1*cd1 + cd2*cd2), 1e-12f);
        float d0 = fd0/nf - cd0/nc;
        float d1 = fd1/nf - cd1/nc;
        float d2 = fd2/nf - cd2/nc;
        acc += d0*d0 + d1*d1 + d2*d2;
    }
    __shared__ float red[256];
    red[t] = acc;
    __syncthreads();
    for (int s = 128; s > 0; s >>= 1) {
        if (t < s) red[t] += red[t + s];
        __syncthreads();
    }
    if (t == 0) partials[blockIdx.x] = red[0];
}

// ---------------- Kernel 5: deterministic final reduction -------------------
__global__ void __launch_bounds__(256) k_final(const float* __restrict__ partials,
                                               float* __restrict__ out) {
    const int t = threadIdx.x;
    float acc = 0.0f;
    for (int i = t; i < 2048; i += 256) acc += partials[i];
    __shared__ float red[256];
    red[t] = acc;
    __syncthreads();
    for (int s = 128; s > 0; s >>= 1) {
        if (t < s) red[t] += red[t + s];
        __syncthreads();
    }
    if (t == 0) out[0] = red[0] / 12582912.0f;   // mean over 16*3*512*512
}

extern "C" void kernel_launch(void* const* d_in, const int* in_sizes, int n_in,
                              void* d_out, int out_size, void* d_ws, size_t ws_size,
                              hipStream_t stream) {
    const float* clear_ = (const float*)d_in[0];
    const float* foggy  = (const float*)d_in[1];
    float* out = (float*)d_out;

    char* ws = (char*)d_ws;
    const size_t plane = (size_t)BSZ * HW * sizeof(float);  // 16 MB
    float* hmin     = (float*)(ws);
    float* dark     = (float*)(ws + plane);
    float* An       = (float*)(ws + 2*plane);
    float* partials = (float*)(ws + 2*plane + 256);

    k_hmin  <<<dim3(H, BSZ), 256, 0, stream>>>(foggy, hmin);
    k_vmin  <<<(BSZ*HW + 255)/256, 256, 0, stream>>>(hmin, dark);
    k_select<<<BSZ, 1024, 0, stream>>>(dark, foggy, An);
    k_loss  <<<(BSZ*HW)/2048, 256, 0, stream>>>(clear_, foggy, An, partials);
    k_final <<<1, 256, 0, stream>>>(partials, out);
}